// RnCLoss_81862076661829
// MI455X (gfx1250) — compile-verified
//
#include <hip/hip_runtime.h>
#include <math.h>

typedef __attribute__((ext_vector_type(2))) float v2f;
typedef __attribute__((ext_vector_type(8))) float v8f;

#define NN 512
#define DD 256
#define TEMP 2.0f

// features laid out (256, 2, 256); logical row i of f = concat(view0, view1):
//   i < 256  -> feat + i*512
//   i >= 256 -> feat + (i-256)*512 + 256
__device__ __forceinline__ const float* frow(const float* feat, int i) {
    return feat + (i & 255) * 512 + (i >> 8) * 256;
}

// ---------------------------------------------------------------- sqnorms
__global__ void sqnorm_kernel(const float* __restrict__ feat,
                              float* __restrict__ sn) {
    int i = blockIdx.x * blockDim.x + threadIdx.x;
    if (i >= NN) return;
    const float* r = frow(feat, i);
    float s = 0.f;
#pragma unroll 4
    for (int d = 0; d < DD; d += 4) {
        float4 v = *(const float4*)(r + d);
        s += v.x * v.x + v.y * v.y + v.z * v.z + v.w * v.w;
    }
    sn[i] = s;
}

// ------------------------------------------- Gram via WMMA -> logits matrix
// One wave computes one 16x16 tile of G = f * f^T using V_WMMA_F32_16X16X4_F32
// (fp32 A/B, fp32 accumulate: exact-precision tensor-pipe matmul), then emits
// logits[i][j] = -sqrt(max(sn_i + sn_j - 2*G_ij, 0)) / TEMP, diagonal forced 0.
__global__ void gram_logits_kernel(const float* __restrict__ feat,
                                   const float* __restrict__ sn,
                                   float* __restrict__ logits) {
    const int wave = (blockIdx.x * blockDim.x + threadIdx.x) >> 5;
    const int lane = threadIdx.x & 31;
    const int tr = wave >> 5;                 // tile row 0..31
    const int tc = wave & 31;                 // tile col 0..31
    const int m  = lane & 15;                 // row (A) / col (B) within tile
    const int khalf = (lane >> 4) << 1;       // lanes 0-15: K=0,1 ; 16-31: K=2,3

    const float* fa = frow(feat, tr * 16 + m);
    const float* fb = frow(feat, tc * 16 + m);   // B = f^T: same row loads

    v8f acc = {};
#pragma unroll 8
    for (int k = 0; k < DD; k += 4) {
        v2f a = *(const v2f*)(fa + k + khalf);
        v2f b = *(const v2f*)(fb + k + khalf);
        // 8 args: (neg_a, A, neg_b, B, c_mod, C, reuse_a, reuse_b)
        acc = __builtin_amdgcn_wmma_f32_16x16x4_f32(
            false, a, false, b, (short)0, acc, false, false);
    }

    // C/D layout: VGPR r -> (M = r + 8*(lane>=16), N = lane & 15)
    const int gj  = tc * 16 + (lane & 15);
    const float snj = sn[gj];
#pragma unroll
    for (int r = 0; r < 8; ++r) {
        int gi = tr * 16 + r + ((lane >> 4) << 3);
        float sq = sn[gi] + snj - 2.0f * acc[r];
        sq = sq > 0.f ? sq : 0.f;
        float lg = -sqrtf(sq) * (1.0f / TEMP);
        if (gi == gj) lg = 0.f;       // exact diagonal => logits_max == 0
        logits[gi * NN + gj] = lg;
    }
}

// ------------------------------------------------------------ per-row loss
// One block per row i. neg[i,k] = sum_{j!=i} [ld_j >= ld_k] * exp(logit_j);
// the j==i term is exactly exp(0)=1 included iff ld_k==0, removed analytically.
__global__ void rowloss_kernel(const float* __restrict__ logits,
                               const float* __restrict__ labels,
                               float* __restrict__ rowsum) {
    __shared__ float s_exp[NN];
    __shared__ float s_ld[NN];
    __shared__ float s_lg[NN];
    __shared__ float red[256];

    const int i = blockIdx.x;
    const int t = threadIdx.x;
    const float labi = labels[i & 255];

    for (int j = t; j < NN; j += 256) {
        float lg = logits[i * NN + j];
        s_lg[j]  = lg;
        s_exp[j] = expf(lg);
        s_ld[j]  = fabsf(labi - labels[j & 255]);
    }
    __syncthreads();

    float acc = 0.f;
    for (int k = t; k < NN; k += 256) {
        if (k == i) continue;
        const float ldk = s_ld[k];
        float neg = 0.f;
        for (int j = 0; j < NN; ++j)            // LDS broadcast reads
            neg += (s_ld[j] >= ldk) ? s_exp[j] : 0.f;
        if (ldk <= 0.f) neg -= 1.0f;            // drop diagonal contribution
        acc += s_lg[k] - logf(neg);
    }

    red[t] = acc;
    __syncthreads();
    for (int s = 128; s > 0; s >>= 1) {
        if (t < s) red[t] += red[t + s];
        __syncthreads();
    }
    if (t == 0) rowsum[i] = red[0];
}

// ------------------------------------------------------------- final reduce
__global__ void finalize_kernel(const float* __restrict__ rowsum,
                                float* __restrict__ out) {
    __shared__ float red[256];
    int t = threadIdx.x;
    red[t] = rowsum[t] + rowsum[t + 256];
    __syncthreads();
    for (int s = 128; s > 0; s >>= 1) {
        if (t < s) red[t] += red[t + s];
        __syncthreads();
    }
    if (t == 0) out[0] = -red[0] / (512.0f * 511.0f);
}

extern "C" void kernel_launch(void* const* d_in, const int* in_sizes, int n_in,
                              void* d_out, int out_size, void* d_ws, size_t ws_size,
                              hipStream_t stream) {
    const float* feat   = (const float*)d_in[0];   // (256,2,256) f32
    const float* labels = (const float*)d_in[1];   // (256,1) f32

    float* logits = (float*)d_ws;          // 512*512 f32 = 1 MB
    float* sn     = logits + NN * NN;      // 512 f32
    float* rowsum = sn + NN;               // 512 f32
    float* out    = (float*)d_out;

    sqnorm_kernel<<<2, 256, 0, stream>>>(feat, sn);
    gram_logits_kernel<<<128, 256, 0, stream>>>(feat, sn, logits);  // 1024 tiles, 8 waves/block
    rowloss_kernel<<<NN, 256, 0, stream>>>(logits, labels, rowsum);
    finalize_kernel<<<1, 256, 0, stream>>>(rowsum, out);
}